// SSIM_69200513073568
// MI455X (gfx1250) — compile-verified
//
#include <hip/hip_runtime.h>
#include <stdint.h>

// ---------------------------------------------------------------------------
// SSIM on MI455X (gfx1250).
//
// Roofline: 67 MB of reads -> ~2.9us HBM floor (entire working set fits the
// 192 MB L2). A scalar separable conv is f32-VALU bound, so both 7-tap
// Gaussian passes run as banded matmuls on V_WMMA_F32_16X16X4_F32 (K 22->24,
// 6 K-steps, full f32 precision). Round-2 restructure: branch-free fragment
// construction (zero-padded LDS tiles + LDS band table), 5 channels per wave
// sharing A-loads and band fragments -> WMMA-dense inner loops.
// ---------------------------------------------------------------------------

typedef float v2f __attribute__((ext_vector_type(2)));
typedef float v8f __attribute__((ext_vector_type(8)));

#define Hdim 512
#define Wdim 512
#define NB   32
#define TILE 32
#define DATA 38            // TILE + 6 (valid haloed data)
#define SROW 48            // padded rows in sx/sy (>= 47 reachable by A)
#define SSTR 41            // padded cols/stride in sx/sy (odd -> bank-clean)
#define TROW 40            // padded rows in T (phase-2 B reads rows 0..39)
#define TSTR 33
#define NTX  (Wdim / TILE) // 16
#define NTY  (Hdim / TILE) // 16
#define NWG  (NTX * NTY * NB) // 8192 partials
#define C1f  0.0001f       // 0.01^2
#define C2f  0.0009f       // 0.03^2

#define WMMA_F32(A, B, C) \
  __builtin_amdgcn_wmma_f32_16x16x4_f32(false, (A), false, (B), (short)0, (C), false, false)

__global__ __launch_bounds__(256) void ssim_tile_kernel(
    const float* __restrict__ x, const float* __restrict__ y,
    const float* __restrict__ k2d, float* __restrict__ partial)
{
  __shared__ float sx[SROW * SSTR];      // x tile, zero-padded
  __shared__ float sy[SROW * SSTR];      // y tile, zero-padded
  __shared__ float T[5 * TROW * TSTR];   // horizontal-pass result, 5 channels
  __shared__ float R[5 * TILE * TSTR];   // final conv results, 5 channels
  __shared__ float gt[64];               // banded gaussian lookup, gt[32+j]=g[j]
  __shared__ float red[256];

  const int tid  = threadIdx.x;
  const int lane = tid & 31;
  const int wave = tid >> 5;

  // Band table: k2d = g g^T (normalized) => g[j] = k2d[j][3] / sqrt(k2d[3][3]).
  if (tid < 64) {
    float g3 = sqrtf(k2d[3 * 7 + 3]);
    int j = tid - 32;
    gt[tid] = (j >= 0 && j <= 6) ? (k2d[j * 7 + 3] / g3) : 0.0f;
  }

  // Zero the pad region once (rows >= 38 or cols >= 38).
  for (int i = tid; i < SROW * SSTR; i += 256) {
    int r = i / SSTR, c = i % SSTR;
    if (r >= DATA || c >= DATA) { sx[i] = 0.0f; sy[i] = 0.0f; }
  }

  const int gx0 = (int)blockIdx.x * TILE - 3;
  const int gy0 = (int)blockIdx.y * TILE - 3;
  const size_t plane = (size_t)blockIdx.z * ((size_t)Hdim * Wdim);
  const float* xb = x + plane;
  const float* yb = y + plane;

  const bool interior =
      (gx0 >= 0) && (gy0 >= 0) && (gx0 + DATA <= Wdim) && (gy0 + DATA <= Hdim);

  if (interior) {
    // gfx1250 async DMA: global -> LDS, no VGPR data path.
    for (int i = tid; i < DATA * DATA; i += 256) {
      int r = i / DATA, c = i % DATA;
      size_t goff = ((size_t)(gy0 + r) * Wdim + (size_t)(gx0 + c)) * sizeof(float);
      uint32_t lx = (uint32_t)(uintptr_t)&sx[r * SSTR + c];
      uint32_t ly = (uint32_t)(uintptr_t)&sy[r * SSTR + c];
      uint64_t ax = (uint64_t)(uintptr_t)xb + goff;
      uint64_t ay = (uint64_t)(uintptr_t)yb + goff;
      asm volatile("global_load_async_to_lds_b32 %0, %1, off" :: "v"(lx), "v"(ax) : "memory");
      asm volatile("global_load_async_to_lds_b32 %0, %1, off" :: "v"(ly), "v"(ay) : "memory");
    }
    asm volatile("s_wait_asynccnt 0" ::: "memory");
  } else {
    // Border tiles: guarded loads; zeros == conv 'SAME' zero padding.
    for (int i = tid; i < DATA * DATA; i += 256) {
      int r = i / DATA, c = i % DATA;
      int gr = gy0 + r, gc = gx0 + c;
      bool ok = (gr >= 0) && (gr < Hdim) && (gc >= 0) && (gc < Wdim);
      size_t goff = (size_t)(ok ? gr : 0) * Wdim + (size_t)(ok ? gc : 0);
      sx[r * SSTR + c] = ok ? xb[goff] : 0.0f;
      sy[r * SSTR + c] = ok ? yb[goff] : 0.0f;
    }
  }
  __syncthreads();

  // Per-lane band fragments (identical values serve as horizontal-pass B and
  // vertical-pass A under the 16x4/4x16 f32 VGPR layouts): frag[e] = g[k - mn].
  const int mn = lane & 15;          // A row / B col owned by this lane
  const int kh = (lane >> 4) * 2;    // K sub-offset of this half-wave
  v2f band[6];
  #pragma unroll
  for (int kk = 0; kk < 6; ++kk) {
    #pragma unroll
    for (int e = 0; e < 2; ++e) {
      int k = kk * 4 + kh + e;
      band[kk][e] = gt[(k - mn) + 32];   // index in [17, 55]
    }
  }

  // ------------------------------------------------------------------
  // Phase 1: horizontal conv. T_blk = In(16x24) x Band(24x16).
  // 6 blocks (3 row x 2 col) -> waves 0..5; each wave does all 5 channels
  // sharing the A-loads and band fragment per K-step.
  // ------------------------------------------------------------------
  if (wave < 6) {
    int r0 = (wave >> 1) * 16, c0 = (wave & 1) * 16;
    v8f a0 = {}, a1 = {}, a2 = {}, a3 = {}, a4 = {};
    #pragma unroll
    for (int kk = 0; kk < 6; ++kk) {
      v2f fx, fy, fxx, fyy, fxy;
      #pragma unroll
      for (int e = 0; e < 2; ++e) {
        int idx = (r0 + mn) * SSTR + (c0 + kk * 4 + kh + e);  // always in-bounds
        float vx = sx[idx];
        float vy = sy[idx];
        fx[e] = vx; fy[e] = vy;
        fxx[e] = vx * vx; fyy[e] = vy * vy; fxy[e] = vx * vy;
      }
      v2f b = band[kk];
      a0 = WMMA_F32(fx,  b, a0);
      a1 = WMMA_F32(fy,  b, a1);
      a2 = WMMA_F32(fxx, b, a2);
      a3 = WMMA_F32(fyy, b, a3);
      a4 = WMMA_F32(fxy, b, a4);
    }
    int rbase = r0 + (lane >> 4) * 8;
    #pragma unroll
    for (int v = 0; v < 8; ++v) {
      int row = rbase + v;
      if (row < TROW) {                  // only trims the rb==2 block
        int o = row * TSTR + (c0 + mn);
        T[0 * TROW * TSTR + o] = a0[v];
        T[1 * TROW * TSTR + o] = a1[v];
        T[2 * TROW * TSTR + o] = a2[v];
        T[3 * TROW * TSTR + o] = a3[v];
        T[4 * TROW * TSTR + o] = a4[v];
      }
    }
  }
  __syncthreads();

  // ------------------------------------------------------------------
  // Phase 2: vertical conv. R_blk = Band(16x24) x T(24x16).
  // 4 blocks (2x2) -> waves 0..3; all 5 channels share the band A-fragment.
  // ------------------------------------------------------------------
  if (wave < 4) {
    int r0 = (wave >> 1) * 16, c0 = (wave & 1) * 16;
    v8f a0 = {}, a1 = {}, a2 = {}, a3 = {}, a4 = {};
    #pragma unroll
    for (int kk = 0; kk < 6; ++kk) {
      v2f b0, b1, b2, b3, b4;
      #pragma unroll
      for (int e = 0; e < 2; ++e) {
        int row = r0 + kk * 4 + kh + e;                 // <= 39 < TROW
        int o = row * TSTR + (c0 + mn);
        b0[e] = T[0 * TROW * TSTR + o];
        b1[e] = T[1 * TROW * TSTR + o];
        b2[e] = T[2 * TROW * TSTR + o];
        b3[e] = T[3 * TROW * TSTR + o];
        b4[e] = T[4 * TROW * TSTR + o];
      }
      v2f a = band[kk];
      a0 = WMMA_F32(a, b0, a0);
      a1 = WMMA_F32(a, b1, a1);
      a2 = WMMA_F32(a, b2, a2);
      a3 = WMMA_F32(a, b3, a3);
      a4 = WMMA_F32(a, b4, a4);
    }
    int rbase = r0 + (lane >> 4) * 8;
    #pragma unroll
    for (int v = 0; v < 8; ++v) {
      int o = (rbase + v) * TSTR + (c0 + mn);           // rows 0..31
      R[0 * TILE * TSTR + o] = a0[v];
      R[1 * TILE * TSTR + o] = a1[v];
      R[2 * TILE * TSTR + o] = a2[v];
      R[3 * TILE * TSTR + o] = a3[v];
      R[4 * TILE * TSTR + o] = a4[v];
    }
  }
  __syncthreads();

  // ------------------------------------------------------------------
  // SSIM map + deterministic fixed-order reduction.
  // ------------------------------------------------------------------
  float lsum = 0.0f;
  #pragma unroll
  for (int p = 0; p < 4; ++p) {
    int pix = tid + p * 256;
    int row = pix >> 5, col = pix & 31;
    int o = row * TSTR + col;
    float mx = R[0 * TILE * TSTR + o];
    float my = R[1 * TILE * TSTR + o];
    float xx = R[2 * TILE * TSTR + o];
    float yy = R[3 * TILE * TSTR + o];
    float xy = R[4 * TILE * TSTR + o];
    float sxx = xx - mx * mx;
    float syy = yy - my * my;
    float sxy = xy - mx * my;
    float num = (2.f * mx * my + C1f) * (2.f * sxy + C2f);
    float den = (mx * mx + my * my + C1f) * (sxx + syy + C2f);
    lsum += num / den;
  }
  red[tid] = lsum;
  __syncthreads();
  for (int s = 128; s > 0; s >>= 1) {
    if (tid < s) red[tid] += red[tid + s];
    __syncthreads();
  }
  if (tid == 0)
    partial[((int)blockIdx.z * NTY + (int)blockIdx.y) * NTX + (int)blockIdx.x] = red[0];
}

// Single-block deterministic final reduction of the 8192 partials.
__global__ __launch_bounds__(256) void ssim_reduce_kernel(
    const float* __restrict__ partial, float* __restrict__ out)
{
  __shared__ float red[256];
  int tid = threadIdx.x;
  float s = 0.0f;
  for (int i = tid; i < NWG; i += 256) s += partial[i];
  red[tid] = s;
  __syncthreads();
  for (int k = 128; k > 0; k >>= 1) {
    if (tid < k) red[tid] += red[tid + k];
    __syncthreads();
  }
  if (tid == 0)
    out[0] = red[0] * (1.0f / (float)((size_t)NB * Hdim * Wdim));
}

extern "C" void kernel_launch(void* const* d_in, const int* in_sizes, int n_in,
                              void* d_out, int out_size, void* d_ws, size_t ws_size,
                              hipStream_t stream) {
  (void)in_sizes; (void)n_in; (void)out_size; (void)ws_size;
  const float* x   = (const float*)d_in[0];
  const float* y   = (const float*)d_in[1];
  const float* k2d = (const float*)d_in[2];
  float* partial   = (float*)d_ws;   // NWG floats = 32 KB of scratch

  dim3 grid(NTX, NTY, NB), block(256);
  ssim_tile_kernel<<<grid, block, 0, stream>>>(x, y, k2d, partial);
  ssim_reduce_kernel<<<1, 256, 0, stream>>>(partial, (float*)d_out);
}